// LightGCN_11261404250193
// MI455X (gfx1250) — compile-verified
//
#include <hip/hip_runtime.h>
#include <hip/hip_bf16.h>

#define N_USERS   200000
#define N_ITEMS   100000
#define N_NODES   300000   // N_USERS + N_ITEMS
#define EMBED_DIM 64
#define N_EDGES   4800000
#define BATCH     16384

typedef __attribute__((ext_vector_type(2))) float v2f;
typedef __attribute__((ext_vector_type(8))) float v8f;

// ---------------------------------------------------------------------------
// A[i] = concat(user_embed, item_embed), float4-vectorized copy.
// ---------------------------------------------------------------------------
__global__ void lgcn_initA(const float4* __restrict__ ue,
                           const float4* __restrict__ ie,
                           float4* __restrict__ A) {
  size_t i = (size_t)blockIdx.x * blockDim.x + threadIdx.x;   // float4 index
  const size_t nu4  = (size_t)N_USERS * (EMBED_DIM / 4);
  const size_t tot4 = (size_t)N_NODES * (EMBED_DIM / 4);
  if (i >= tot4) return;
  A[i] = (i < nu4) ? ue[i] : ie[i - nu4];
}

__global__ void lgcn_zero(float4* __restrict__ B) {
  size_t i = (size_t)blockIdx.x * blockDim.x + threadIdx.x;
  const size_t tot4 = (size_t)N_NODES * (EMBED_DIM / 4);
  if (i >= tot4) return;
  B[i] = make_float4(0.f, 0.f, 0.f, 0.f);
}

// ---------------------------------------------------------------------------
// Layer-0 term: out_u[b] = user_embed[users[b]], out_i[b] = item_embed[items[b]]
// (also initializes d_out, which is poisoned by the harness).
// ---------------------------------------------------------------------------
__global__ void lgcn_gather_init(const float4* __restrict__ ue,
                                 const float4* __restrict__ ie,
                                 const int* __restrict__ users,
                                 const int* __restrict__ items,
                                 float4* __restrict__ uout,
                                 float4* __restrict__ iout) {
  int idx = blockIdx.x * blockDim.x + threadIdx.x;   // BATCH*16 float4 slots
  if (idx >= BATCH * (EMBED_DIM / 4)) return;
  int b = idx >> 4;
  int q = idx & 15;
  uout[idx] = ue[(size_t)users[b] * 16 + q];
  iout[idx] = ie[(size_t)items[b] * 16 + q];
}

// ---------------------------------------------------------------------------
// SpMM scatter: B[dst] += val * A[src].  16 lanes per edge, float4 per lane,
// hardware f32 atomic adds (L2-resident destination table).
// ---------------------------------------------------------------------------
__global__ void lgcn_scatter(const float* __restrict__ A,
                             float* __restrict__ B,
                             const int* __restrict__ src,
                             const int* __restrict__ dst,
                             const float* __restrict__ val) {
  long long tid = (long long)blockIdx.x * blockDim.x + threadIdx.x;
  if (tid >= (long long)N_EDGES * 16) return;
  int e = (int)(tid >> 4);
  int q = ((int)tid & 15) << 2;                 // float column 0,4,...,60
  int s = src[e];
  int d = dst[e];
  float v = val[e];
  const float4 x = *(const float4*)(A + (size_t)s * EMBED_DIM + q);
  float* out = B + (size_t)d * EMBED_DIM + q;
  unsafeAtomicAdd(out + 0, v * x.x);
  unsafeAtomicAdd(out + 1, v * x.y);
  unsafeAtomicAdd(out + 2, v * x.z);
  unsafeAtomicAdd(out + 3, v * x.w);
}

// ---------------------------------------------------------------------------
// Per-layer batch accumulation: out_u[b] += B[users[b]], out_i[b] += B[N_USERS+items[b]]
// ---------------------------------------------------------------------------
__global__ void lgcn_gather_acc(const float4* __restrict__ B,
                                const int* __restrict__ users,
                                const int* __restrict__ items,
                                float4* __restrict__ uout,
                                float4* __restrict__ iout) {
  int idx = blockIdx.x * blockDim.x + threadIdx.x;
  if (idx >= BATCH * (EMBED_DIM / 4)) return;
  int b = idx >> 4;
  int q = idx & 15;
  float4 u  = B[(size_t)users[b] * 16 + q];
  float4 it = B[(size_t)(N_USERS + items[b]) * 16 + q];
  float4 au = uout[idx];
  float4 ai = iout[idx];
  au.x += u.x;  au.y += u.y;  au.z += u.z;  au.w += u.w;
  ai.x += it.x; ai.y += it.y; ai.z += it.z; ai.w += it.w;
  uout[idx] = au;
  iout[idx] = ai;
}

// ---------------------------------------------------------------------------
// Finalize: one wave32 per 16 (user,item) pairs.
//  1) Gram tile D = U_tile(16x64) x I_tile(64x16)^T via 16 chained
//     V_WMMA_F32_16X16X4_F32 ops (raw, unscaled values).
//  2) Scale the 16 u-rows and 16 i-rows in d_out by 1/4 (layer mean).
//  3) scores = diag(D) * 1/16  (== (u/4)·(i/4)).
// A-frag 16x4: lane<16 -> {K=k0,k0+1}, lane>=16 -> {K=k0+2,k0+3}; B-frag is
// the symmetric 4x16 layout, so both load the same (row = lane&15) pattern.
// ---------------------------------------------------------------------------
__global__ void lgcn_finalize(float* __restrict__ uout,
                              float* __restrict__ iout,
                              float* __restrict__ scores) {
  const int lane  = threadIdx.x & 31;
  const int wave  = threadIdx.x >> 5;
  const int group = blockIdx.x * (blockDim.x >> 5) + wave;   // 0..1023
  const int p0    = group * 16;
  const int mrow  = lane & 15;
  const int koff  = (lane >> 4) * 2;

  const float* Urow = uout + (size_t)(p0 + mrow) * EMBED_DIM;
  const float* Irow = iout + (size_t)(p0 + mrow) * EMBED_DIM;

  v8f c = {};
#pragma unroll
  for (int k0 = 0; k0 < EMBED_DIM; k0 += 4) {
    v2f a = *(const v2f*)(Urow + k0 + koff);
    v2f b = *(const v2f*)(Irow + k0 + koff);
    c = __builtin_amdgcn_wmma_f32_16x16x4_f32(
        /*neg_a=*/false, a, /*neg_b=*/false, b,
        /*c_mod=*/(short)0, c, /*reuse_a=*/false, /*reuse_b=*/false);
  }

  // Scale embeddings by 1/4 and write back (after raw fragment loads above).
  {
    const int colbase = (lane >> 4) * 32;
    float4* u4 = (float4*)(uout + (size_t)(p0 + mrow) * EMBED_DIM + colbase);
    float4* i4 = (float4*)(iout + (size_t)(p0 + mrow) * EMBED_DIM + colbase);
#pragma unroll
    for (int t = 0; t < 8; ++t) {
      float4 uu = u4[t];
      uu.x *= 0.25f; uu.y *= 0.25f; uu.z *= 0.25f; uu.w *= 0.25f;
      u4[t] = uu;
      float4 ii = i4[t];
      ii.x *= 0.25f; ii.y *= 0.25f; ii.z *= 0.25f; ii.w *= 0.25f;
      i4[t] = ii;
    }
  }

  // Diagonal extraction: m<8 -> VGPR m on lane m; m>=8 -> VGPR m-8 on lane m+16.
  int r = -1, m = 0;
  if (lane < 8)        { r = lane;      m = lane;      }
  else if (lane >= 24) { r = lane - 24; m = lane - 16; }
  if (r >= 0) {
    float diag = c[0];
#pragma unroll
    for (int t = 1; t < 8; ++t)
      if (r == t) diag = c[t];
    scores[p0 + m] = diag * 0.0625f;   // (1/4)*(1/4)
  }
}

// ---------------------------------------------------------------------------
extern "C" void kernel_launch(void* const* d_in, const int* in_sizes, int n_in,
                              void* d_out, int out_size, void* d_ws, size_t ws_size,
                              hipStream_t stream) {
  (void)in_sizes; (void)n_in; (void)out_size; (void)ws_size;

  const float* ue   = (const float*)d_in[0];
  const float* ie   = (const float*)d_in[1];
  const int*   esrc = (const int*)d_in[2];
  const int*   edst = (const int*)d_in[3];
  const float* eval = (const float*)d_in[4];
  const int*   users = (const int*)d_in[5];
  const int*   items = (const int*)d_in[6];

  float* uout   = (float*)d_out;                        // [BATCH,64]
  float* iout   = uout + (size_t)BATCH * EMBED_DIM;     // [BATCH,64]
  float* scores = iout + (size_t)BATCH * EMBED_DIM;     // [BATCH]

  float* A = (float*)d_ws;                              // [N_NODES,64]
  float* B = A + (size_t)N_NODES * EMBED_DIM;           // [N_NODES,64]

  const int T = 256;
  const size_t node4 = (size_t)N_NODES * (EMBED_DIM / 4);
  const int batch4 = BATCH * (EMBED_DIM / 4);

  lgcn_initA<<<(unsigned)((node4 + T - 1) / T), T, 0, stream>>>(
      (const float4*)ue, (const float4*)ie, (float4*)A);
  lgcn_gather_init<<<(batch4 + T - 1) / T, T, 0, stream>>>(
      (const float4*)ue, (const float4*)ie, users, items,
      (float4*)uout, (float4*)iout);

  for (int layer = 0; layer < 3; ++layer) {
    lgcn_zero<<<(unsigned)((node4 + T - 1) / T), T, 0, stream>>>((float4*)B);
    const long long nthr = (long long)N_EDGES * 16;
    lgcn_scatter<<<(unsigned)((nthr + T - 1) / T), T, 0, stream>>>(
        A, B, esrc, edst, eval);
    lgcn_gather_acc<<<(batch4 + T - 1) / T, T, 0, stream>>>(
        (const float4*)B, users, items, (float4*)uout, (float4*)iout);
    float* tmp = A; A = B; B = tmp;
  }

  // 1024 wave-groups of 16 pairs; 8 waves (256 threads) per block -> 128 blocks.
  lgcn_finalize<<<BATCH / 16 / 8, 256, 0, stream>>>(uout, iout, scores);
}